// MILoss_10814727652063
// MI455X (gfx1250) — compile-verified
//
#include <hip/hip_runtime.h>
#include <hip/hip_bf16.h>

typedef __attribute__((ext_vector_type(2))) float v2f;
typedef __attribute__((ext_vector_type(8))) float v8f;

#define NUM_TASKS   8
#define NUM_EXPERTS 64
#define NBATCH      32
#define SEQ         8192
#define NTOK        (NBATCH * SEQ)   // 262144 tokens
#define NGRP        (NTOK / 4)       // 65536 groups of 4 tokens
#define TOPK        2
#define WMI_C       0.01f
#define EPS_C       0.0001f

// ---------------------------------------------------------------------------
// DPP-based 16-lane reductions (pure VALU, no LDS, no dscnt waits).
// A DPP "row" is 16 lanes == one token-half of the wave, so the two halves
// reduce independently, which is exactly what the WMMA B layout needs.
// ---------------------------------------------------------------------------
template <int CTRL>
__device__ __forceinline__ float dpp_movf(float x) {
    return __int_as_float(__builtin_amdgcn_update_dpp(
        0, __float_as_int(x), CTRL, 0xF, 0xF, true));
}

__device__ __forceinline__ float row16_max(float x) {
    x = fmaxf(x, dpp_movf<0xB1>(x));   // quad_perm(1,0,3,2): xor-1 within quad
    x = fmaxf(x, dpp_movf<0x4E>(x));   // quad_perm(2,3,0,1): xor-2 -> quad total
    x = fmaxf(x, dpp_movf<0x141>(x));  // row_half_mirror: other quad within 8
    x = fmaxf(x, dpp_movf<0x140>(x));  // row_mirror: other 8-half within 16
    return x;
}

__device__ __forceinline__ float row16_sum(float x) {
    x += dpp_movf<0xB1>(x);
    x += dpp_movf<0x4E>(x);
    x += dpp_movf<0x141>(x);
    x += dpp_movf<0x140>(x);
    return x;
}

// ---------------------------------------------------------------------------
// Kernel 1: fused softmax + one-hot segment-GEMM via V_WMMA_F32_16X16X4_F32.
// Each wave: 4 tokens/iteration. A = 16x4 one-hot(task)*rcp(sum) f32 (softmax
// normalization folded into A since seg = A x B is bilinear and A has one
// nonzero per K column), B = 4x16 raw exp(x - max) f32 (4 expert tiles),
// C accumulates seg[16 tasks][64 experts] in 4x v8f.
// Layout (ISA 7.12.2): lane half h <-> tokens {2h, 2h+1}; VGPR v <-> token
// 2h+v; C/D VGPR i <-> task i (lanes 0-15) / i+8 (lanes 16-31, zero here).
// ---------------------------------------------------------------------------
__global__ __launch_bounds__(256)
void miloss_seg_kernel(const float* __restrict__ logits,
                       const int*   __restrict__ labels,
                       float*       __restrict__ gseg /* [8*64] */) {
    __shared__ float sseg[NUM_TASKS * NUM_EXPERTS];
    for (int t = threadIdx.x; t < NUM_TASKS * NUM_EXPERTS; t += blockDim.x)
        sseg[t] = 0.0f;
    __syncthreads();

    const int lane = threadIdx.x & 31;
    const int h    = lane >> 4;   // wave half: tokens 2h, 2h+1
    const int c    = lane & 15;   // column within 16-wide expert tile
    const int wavesPerBlk = blockDim.x >> 5;
    const int wid    = blockIdx.x * wavesPerBlk + (threadIdx.x >> 5);
    const int nwaves = gridDim.x * wavesPerBlk;

    v8f acc0 = {}, acc1 = {}, acc2 = {}, acc3 = {};

    for (int g = wid; g < NGRP; g += nwaves) {
        const int tg  = g << 2;                 // first token of group
        const int lab = labels[tg >> 13];       // 4 tokens never straddle a batch

        // token 2h : experts c, c+16, c+32, c+48
        const float* p0 = logits + (size_t)(tg + 2 * h) * NUM_EXPERTS + c;
        float x00 = p0[0], x01 = p0[16], x02 = p0[32], x03 = p0[48];
        // token 2h+1
        const float* p1 = p0 + NUM_EXPERTS;
        float x10 = p1[0], x11 = p1[16], x12 = p1[32], x13 = p1[48];

        // softmax max over the token's 64 values (4 regs x 16 lanes of a half)
        float m0 = row16_max(fmaxf(fmaxf(x00, x01), fmaxf(x02, x03)));
        float m1 = row16_max(fmaxf(fmaxf(x10, x11), fmaxf(x12, x13)));

        float e00 = __expf(x00 - m0), e01 = __expf(x01 - m0);
        float e02 = __expf(x02 - m0), e03 = __expf(x03 - m0);
        float e10 = __expf(x10 - m1), e11 = __expf(x11 - m1);
        float e12 = __expf(x12 - m1), e13 = __expf(x13 - m1);

        float s0 = row16_sum((e00 + e01) + (e02 + e03));
        float s1 = row16_sum((e10 + e11) + (e12 + e13));

        // fast rcp (v_rcp_f32): softmax normalization, ~1 ulp is plenty here
        const float r0 = __builtin_amdgcn_rcpf(s0);
        const float r1 = __builtin_amdgcn_rcpf(s1);

        // B operands: raw exp values; element v of the v2f = VGPR v = token 2h+v
        v2f b0, b1, b2, b3;
        b0[0] = e00; b0[1] = e10;
        b1[0] = e01; b1[1] = e11;
        b2[0] = e02; b2[1] = e12;
        b3[0] = e03; b3[1] = e13;

        // A operand: one-hot row (M = lane&15) scaled by the token's 1/sum
        const float sel = (lab == c) ? 1.0f : 0.0f;
        v2f a; a[0] = sel * r0; a[1] = sel * r1;

        // D = A x B + C, full f32 (16x16x4)
        acc0 = __builtin_amdgcn_wmma_f32_16x16x4_f32(false, a, false, b0,
                                                     (short)0, acc0, false, false);
        acc1 = __builtin_amdgcn_wmma_f32_16x16x4_f32(false, a, false, b1,
                                                     (short)0, acc1, false, false);
        acc2 = __builtin_amdgcn_wmma_f32_16x16x4_f32(false, a, false, b2,
                                                     (short)0, acc2, false, false);
        acc3 = __builtin_amdgcn_wmma_f32_16x16x4_f32(false, a, false, b3,
                                                     (short)0, acc3, false, false);
    }

    // Wave epilogue: lanes 0-15 hold tasks 0-7 in acc element i (lanes 16-31
    // hold task rows 8-15 which are identically zero -> skip).
    if (lane < 16) {
#pragma unroll
        for (int i = 0; i < NUM_TASKS; ++i) {
            atomicAdd(&sseg[i * NUM_EXPERTS +  0 + c], acc0[i]);
            atomicAdd(&sseg[i * NUM_EXPERTS + 16 + c], acc1[i]);
            atomicAdd(&sseg[i * NUM_EXPERTS + 32 + c], acc2[i]);
            atomicAdd(&sseg[i * NUM_EXPERTS + 48 + c], acc3[i]);
        }
    }
    __syncthreads();

    for (int t = threadIdx.x; t < NUM_TASKS * NUM_EXPERTS; t += blockDim.x)
        atomicAdd(&gseg[t], sseg[t]);
}

// ---------------------------------------------------------------------------
// Kernel 2: tiny epilogue on the 8x64 table -> scalar loss.
// ---------------------------------------------------------------------------
__global__ __launch_bounds__(512)
void miloss_final_kernel(const float* __restrict__ gseg,
                         const int*   __restrict__ labels,
                         float*       __restrict__ out) {
    __shared__ float red[NUM_TASKS * NUM_EXPERTS];
    __shared__ float counts[NUM_TASKS];
    __shared__ float pti[NUM_TASKS];
    __shared__ float pei[NUM_EXPERTS];
    __shared__ float stot;

    const int t = threadIdx.x;  // 512 threads, one per (task, expert)

    if (t < NUM_TASKS) {
        int cnt = 0;
        for (int b = 0; b < NBATCH; ++b) cnt += (labels[b] == t) ? 1 : 0;
        counts[t] = (float)cnt * (float)SEQ;
    }
    __syncthreads();

    const float pre = gseg[t] * counts[t >> 6];

    // tot = sum(pre) / TOPK
    red[t] = pre;
    __syncthreads();
#pragma unroll
    for (int s = 256; s > 0; s >>= 1) {
        if (t < s) red[t] += red[t + s];
        __syncthreads();
    }
    if (t == 0) stot = red[0] / (float)TOPK + EPS_C;
    __syncthreads();

    const float gate = pre / stot;

    red[t] = gate;
    __syncthreads();
    if (t < NUM_TASKS) {
        float s = 0.0f;
        for (int e = 0; e < NUM_EXPERTS; ++e) s += red[t * NUM_EXPERTS + e];
        pti[t] = s + EPS_C;
    }
    if (t < NUM_EXPERTS) {
        float s = 0.0f;
        for (int k = 0; k < NUM_TASKS; ++k) s += red[k * NUM_EXPERTS + t];
        pei[t] = s + EPS_C;
    }
    __syncthreads();

    const float term = -gate * __logf(gate / pti[t >> 6] / pei[t & 63] + EPS_C);
    __syncthreads();
    red[t] = term;
    __syncthreads();
#pragma unroll
    for (int s = 256; s > 0; s >>= 1) {
        if (t < s) red[t] += red[t + s];
        __syncthreads();
    }
    if (t == 0) out[0] = WMI_C * red[0];
}

// ---------------------------------------------------------------------------
extern "C" void kernel_launch(void* const* d_in, const int* in_sizes, int n_in,
                              void* d_out, int out_size, void* d_ws, size_t ws_size,
                              hipStream_t stream) {
    const float* logits = (const float*)d_in[0];
    const int*   labels = (const int*)d_in[1];
    float*       gseg   = (float*)d_ws;              // 512 floats of scratch
    float*       out    = (float*)d_out;

    // zero the global segment-sum table (graph-capturable memset node)
    hipMemsetAsync(gseg, 0, NUM_TASKS * NUM_EXPERTS * sizeof(float), stream);

    // 512 blocks x 8 waves = 4096 waves, 16 token-groups each: enough
    // in-flight 64B segments to track the 23.3 TB/s HBM floor (~2.75 us).
    miloss_seg_kernel<<<512, 256, 0, stream>>>(logits, labels, gseg);
    miloss_final_kernel<<<1, 512, 0, stream>>>(gseg, labels, out);
}